// CRF_755914244123
// MI455X (gfx1250) — compile-verified
//
#include <hip/hip_runtime.h>
#include <hip/hip_bf16.h>
#include <stdint.h>

// Problem constants (from reference): B=64, T=1024, C=256
#define Bn   64
#define Tn   1024
#define Cn   256
#define TM1  (Tn - 1)

#define AS3 __attribute__((address_space(3)))
#define AS1 __attribute__((address_space(1)))

// ---------------------------------------------------------------------------
// Kernel 1: Viterbi forward recursion.
// One workgroup per batch element. 1024 threads = 32 waves (8 per SIMD32).
// LDS: full 256KB fp32 transition matrix + alpha[256] + 4x256 partial (val,idx).
// Thread (c, sub): reduces p in [sub*64, sub*64+64) for output class c.
// Backpointers stored as uint8 into d_out's bytes (scratch phase).
// ---------------------------------------------------------------------------
__global__ __launch_bounds__(1024, 1)
void viterbi_forward(const float* __restrict__ seq,
                     const float* __restrict__ trans,
                     unsigned char* __restrict__ bp,
                     float* __restrict__ alpha_out)
{
    extern __shared__ float smem[];
    float* tr    = smem;              // 65536 floats (256KB)
    float* alpha = smem + 65536;      // 256 floats
    float* pv    = alpha + Cn;        // 1024 floats
    int*   pi    = (int*)(pv + 1024); // 1024 ints

    const int b   = blockIdx.x;
    const int tid = threadIdx.x;
    const int c   = tid & (Cn - 1);
    const int sub = tid >> 8;            // 0..3
    const int p0  = sub * (Cn / 4);      // 64-wide p slice

    // ---- Stage transitions (256x256 f32, row-major) into LDS via TDM ----
#if __has_builtin(__builtin_amdgcn_tensor_load_to_lds)
    {
        typedef unsigned int u32x4 __attribute__((ext_vector_type(4)));
        typedef int          i32x8 __attribute__((ext_vector_type(8)));
        typedef int          i32x4 __attribute__((ext_vector_type(4)));
        if (tid < 32) { // one wave issues the TDM DMA (EXEC ignored by TDM)
            unsigned long long ga = (unsigned long long)(uintptr_t)trans;
            unsigned int lds_addr =
                (unsigned int)(unsigned long long)(AS3 float*)tr;
            // D# group0: count=1, lds_addr, global_addr[56:0], type=2
            u32x4 g0 = { 1u,
                         lds_addr,
                         (unsigned int)(ga & 0xFFFFFFFFu),
                         (unsigned int)((ga >> 32) & 0x1FFFFFFu) | (2u << 30) };
            // D# group1: data_size=4B(code 2); tensor 256x256; tile 256x256;
            // dim0_stride=256; dim1_stride=65536
            i32x8 g1 = { 0x00020000,          // wg_mask=0, data_size=2
                         (int)(256u << 16),   // tensor_dim0[15:0]=256
                         (int)(256u << 16),   // tensor_dim1[15:0]=256
                         (int)(256u << 16),   // tile_dim0=256
                         0x00000100,          // tile_dim1=256, tile_dim2=0
                         256,                 // tensor_dim0_stride[31:0]
                         0,                   // stride hi / dim1_stride lo
                         1 };                 // tensor_dim1_stride[47:16]=1 (65536)
            i32x4 g2 = { 0, 0, 0, 0 };
            i32x4 g3 = { 0, 0, 0, 0 };
            i32x8 g4 = { 0, 0, 0, 0, 0, 0, 0, 0 };
            __builtin_amdgcn_tensor_load_to_lds(g0, g1, g2, g3, g4, 0);
#if __has_builtin(__builtin_amdgcn_s_wait_tensorcnt)
            __builtin_amdgcn_s_wait_tensorcnt(0);
#endif
        }
    }
#else
    {
        const float4* tsrc = (const float4*)trans;
        float4*       tdst = (float4*)tr;
        for (int i = tid; i < (Cn * Cn) / 4; i += 1024) tdst[i] = tsrc[i];
    }
#endif

    const float* seqb = seq + (size_t)b * Tn * Cn;
    if (tid < Cn) alpha[tid] = seqb[tid];   // alpha_0 = emissions[0]
    __syncthreads();

    unsigned char* bpb = bp + (size_t)b * TM1 * Cn;

    for (int t = 1; t < Tn; ++t) {
        float emit = 0.f;
        if (sub == 0) {
            emit = seqb[t * Cn + c];        // issued early, consumed after barrier
            if (t + 1 < Tn) __builtin_prefetch(&seqb[(t + 1) * Cn + c], 0, 0);
        }

        // max-plus partial reduction over this thread's 64-wide p slice
        float best = -__builtin_inff();
        int   bidx = p0;
        #pragma unroll
        for (int pp = 0; pp < Cn / 4; pp += 4) {
            const float4 av = *(const float4*)&alpha[p0 + pp];  // ds_load_b128
            float m0 = av.x + tr[(p0 + pp + 0) * Cn + c];
            float m1 = av.y + tr[(p0 + pp + 1) * Cn + c];
            float m2 = av.z + tr[(p0 + pp + 2) * Cn + c];
            float m3 = av.w + tr[(p0 + pp + 3) * Cn + c];
            if (m0 > best) { best = m0; bidx = p0 + pp + 0; }
            if (m1 > best) { best = m1; bidx = p0 + pp + 1; }
            if (m2 > best) { best = m2; bidx = p0 + pp + 2; }
            if (m3 > best) { best = m3; bidx = p0 + pp + 3; }
        }
        pv[tid] = best;
        pi[tid] = bidx;
        __syncthreads();

        if (sub == 0) {
            // combine 4 partials in ascending sub order with strict '>' so the
            // first occurrence of the max wins (matches jnp.argmax semantics)
            float bv = pv[c];
            int   bi = pi[c];
            #pragma unroll
            for (int s = 1; s < 4; ++s) {
                float v = pv[s * Cn + c];
                int   j = pi[s * Cn + c];
                if (v > bv) { bv = v; bi = j; }
            }
            alpha[c] = bv + emit;
            bpb[(t - 1) * Cn + c] = (unsigned char)bi;
        }
        __syncthreads();
    }

    if (sub == 0) alpha_out[b * Cn + c] = alpha[c];
}

// ---------------------------------------------------------------------------
// Kernel 2: backtrace. One block per batch element; stage the 256KB of this
// batch's backpointers into LDS (bulk, async if available), then a single
// lane walks the 1023-step dependent chain at LDS latency.
// ---------------------------------------------------------------------------
__global__ __launch_bounds__(256, 1)
void viterbi_backtrace(const unsigned char* __restrict__ bp,
                       const float* __restrict__ alpha_fin,
                       int* __restrict__ tags)
{
    extern __shared__ unsigned char lbp[];   // TM1*Cn = 261888 bytes
    const int b   = blockIdx.x;
    const int tid = threadIdx.x;
    const unsigned char* bpb = bp + (size_t)b * TM1 * Cn;
    const int n16 = (TM1 * Cn) / 16;         // 16368 16-byte chunks

#if __has_builtin(__builtin_amdgcn_global_load_async_to_lds_b128)
    {
        typedef int v4i __attribute__((vector_size(16)));
        for (int i = tid; i < n16; i += 256) {
            __builtin_amdgcn_global_load_async_to_lds_b128(
                (AS1 v4i*)(bpb + (size_t)i * 16),
                (AS3 v4i*)(lbp + (size_t)i * 16),
                0, 0);
        }
#if __has_builtin(__builtin_amdgcn_s_wait_asynccnt)
        __builtin_amdgcn_s_wait_asynccnt(0);
#endif
    }
#else
    {
        const uint4* src = (const uint4*)bpb;
        uint4*       dst = (uint4*)lbp;
        for (int i = tid; i < n16; i += 256) dst[i] = src[i];
    }
#endif
    __syncthreads();

    if (tid == 0) {
        const float* a = alpha_fin + b * Cn;
        float best = -__builtin_inff();
        int   tag  = 0;
        for (int p = 0; p < Cn; ++p) {
            float v = a[p];
            if (v > best) { best = v; tag = p; }   // first-occurrence argmax
        }
        int* tb = tags + b * Tn;
        tb[Tn - 1] = tag;
        for (int t = TM1 - 1; t >= 0; --t) {
            tag = (int)lbp[t * Cn + tag];
            tb[t] = tag;
        }
    }
}

// ---------------------------------------------------------------------------
// Kernel 3: one-hot expansion. One block per (b,t) row; writes 256 fp32.
// Fully overwrites d_out (which was used as bp byte scratch).
// ---------------------------------------------------------------------------
__global__ __launch_bounds__(256)
void onehot_write(const int* __restrict__ tags, float* __restrict__ out)
{
    const int row = blockIdx.x;              // 0 .. B*T-1
    const int tag = tags[row];
    out[(size_t)row * Cn + threadIdx.x] = (threadIdx.x == tag) ? 1.0f : 0.0f;
}

// ---------------------------------------------------------------------------
extern "C" void kernel_launch(void* const* d_in, const int* in_sizes, int n_in,
                              void* d_out, int out_size, void* d_ws, size_t ws_size,
                              hipStream_t stream) {
    const float* seq   = (const float*)d_in[0];   // [B,T,C] f32
    const float* trans = (const float*)d_in[1];   // [C,C]   f32
    float* out = (float*)d_out;                   // [B,T,C] f32

    // d_out bytes double as backpointer scratch (16.76MB of the 64MB buffer);
    // kernel 3 fully rewrites d_out afterwards (stream-ordered).
    unsigned char* bp = (unsigned char*)d_out;

    // workspace: final alpha (64KB) + tags (256KB) = 320KB
    float* alpha_fin = (float*)d_ws;
    int*   tags      = (int*)((char*)d_ws + (size_t)Bn * Cn * sizeof(float));

    const int shm1 = (65536 + Cn + 1024) * (int)sizeof(float) + 1024 * (int)sizeof(int);
    const int shm2 = TM1 * Cn;  // 261888 bytes

    (void)hipFuncSetAttribute(reinterpret_cast<const void*>(&viterbi_forward),
                              hipFuncAttributeMaxDynamicSharedMemorySize, shm1);
    (void)hipFuncSetAttribute(reinterpret_cast<const void*>(&viterbi_backtrace),
                              hipFuncAttributeMaxDynamicSharedMemorySize, shm2);

    viterbi_forward<<<Bn, 1024, shm1, stream>>>(seq, trans, bp, alpha_fin);
    viterbi_backtrace<<<Bn, 256, shm2, stream>>>(bp, alpha_fin, tags);
    onehot_write<<<Bn * Tn, 256, 0, stream>>>(tags, out);
}